// Underline_86234353369244
// MI455X (gfx1250) — compile-verified
//
#include <hip/hip_runtime.h>
#include <cstdint>

// Problem geometry (fixed by setup_inputs): [B=64, C=3, H=128, W=2048] f32.
#define H_      128
#define W_      2048
#define PLANE_  (H_ * W_)
#define TILE_PX 1024   // pixels per block (256 threads x 4 px)

// ---------- CDNA5 async LDS<->global streaming (gfx1250) ----------
__device__ __forceinline__ void async_load_b128(unsigned lds_off, const float* g) {
    asm volatile("global_load_async_to_lds_b128 %0, %1, off"
                 :: "v"(lds_off), "v"((unsigned long long)(uintptr_t)g)
                 : "memory");
}
__device__ __forceinline__ void async_store_b128(float* g, unsigned lds_off) {
    asm volatile("global_store_async_from_lds_b128 %0, %1, off"
                 :: "v"((unsigned long long)(uintptr_t)g), "v"(lds_off)
                 : "memory");
}
__device__ __forceinline__ void wait_async_zero() {
    asm volatile("s_wait_asynccnt 0x0" ::: "memory");
}

// ---------- kernel 1 (placed first so the disasm snippet shows it):
// fused streaming copy + bbox reduction.
// Grid: (W/TILE_PX, H, B); block: 256 threads, each owns 4 consecutive pixels
// of one row across all 3 channels. Tile is staged global->LDS with the async
// engine, copied out LDS->global asynchronously (overlapping the VALU
// reduction), and the gray/bbox reduction reads the tile from LDS.
__global__ void __launch_bounds__(256)
uline_bbox_copy(const float* __restrict__ in, float* __restrict__ out,
                int* __restrict__ ws) {
    __shared__ __align__(16) float tile[3 * TILE_PX];   // 12 KB
    __shared__ int s_mn, s_mx;

    const int t  = threadIdx.x;
    const int h  = blockIdx.y;
    const int b  = blockIdx.z;
    const int xt = blockIdx.x * TILE_PX;

    if (t == 0) { s_mn = 0x7fffffff; s_mx = -1; }

    const size_t imgbase = (size_t)b * 3 * PLANE_;
    const size_t rowoff  = (size_t)h * W_ + (size_t)xt + (size_t)t * 4;

    // Stage: global -> LDS, 16 B per thread per channel (async engine).
    #pragma unroll
    for (int c = 0; c < 3; ++c) {
        unsigned lo = (unsigned)(uintptr_t)&tile[c * TILE_PX + t * 4];
        async_load_b128(lo, in + imgbase + (size_t)c * PLANE_ + rowoff);
    }
    wait_async_zero();      // wave's own async loads complete
    __syncthreads();        // all waves' loads visible; s_mn/s_mx init visible

    // Copy-out: LDS -> global, issued before the reduction so the async
    // engine streams the store while VALU does the bbox math.
    #pragma unroll
    for (int c = 0; c < 3; ++c) {
        unsigned lo = (unsigned)(uintptr_t)&tile[c * TILE_PX + t * 4];
        async_store_b128(out + imgbase + (size_t)c * PLANE_ + rowoff, lo);
    }

    // Gray (ITU-R 601-2) + per-thread bbox candidates for 4 pixels.
    const float4 r4 = *reinterpret_cast<const float4*>(&tile[0 * TILE_PX + t * 4]);
    const float4 g4 = *reinterpret_cast<const float4*>(&tile[1 * TILE_PX + t * 4]);
    const float4 b4 = *reinterpret_cast<const float4*>(&tile[2 * TILE_PX + t * 4]);

    const int xb = xt + t * 4;
    int mn = 0x7fffffff, mx = -1;
    {
        float gy;
        gy = fmaf(r4.x, 0.2989f, fmaf(g4.x, 0.587f, b4.x * 0.114f));
        if (gy < 0.5f) { mn = min(mn, xb + 0); mx = max(mx, xb + 0); }
        gy = fmaf(r4.y, 0.2989f, fmaf(g4.y, 0.587f, b4.y * 0.114f));
        if (gy < 0.5f) { mn = min(mn, xb + 1); mx = max(mx, xb + 1); }
        gy = fmaf(r4.z, 0.2989f, fmaf(g4.z, 0.587f, b4.z * 0.114f));
        if (gy < 0.5f) { mn = min(mn, xb + 2); mx = max(mx, xb + 2); }
        gy = fmaf(r4.w, 0.2989f, fmaf(g4.w, 0.587f, b4.w * 0.114f));
        if (gy < 0.5f) { mn = min(mn, xb + 3); mx = max(mx, xb + 3); }
    }

    // wave32 butterfly min/max reduction (5 steps).
    #pragma unroll
    for (int o = 16; o > 0; o >>= 1) {
        mn = min(mn, __shfl_xor(mn, o, 32));
        mx = max(mx, __shfl_xor(mx, o, 32));
    }
    if ((t & 31) == 0 && mx >= 0) {     // one wave leader per wave
        atomicMin(&s_mn, mn);
        atomicMax(&s_mx, mx);
    }
    __syncthreads();

    if (t == 0 && s_mx >= 0) {          // <= 3 global atomics per block
        atomicMax(&ws[4 * b + 0], h);
        atomicMin(&ws[4 * b + 1], s_mn);
        atomicMax(&ws[4 * b + 2], s_mx);
    }
    // implicit S_WAIT_IDLE at S_ENDPGM drains the pending async stores
}

// ---------- kernel 0: init per-image bbox state in workspace ----------
// ws layout: per image 4 ints {y1, x0, x1, pad}; y1=-1 means "no black pixel".
__global__ void uline_init_ws(int* __restrict__ ws, int B) {
    int i = blockIdx.x * blockDim.x + threadIdx.x;
    if (i < B) {
        ws[4 * i + 0] = -1;          // y1 = max row  (init -1)
        ws[4 * i + 1] = W_;          // x0 = min col  (init W)
        ws[4 * i + 2] = -1;          // x1 = max col  (init -1)
        ws[4 * i + 3] = 0;
    }
}

// ---------- kernel 2: zero the 3-px band [y1-2..y1] x [x0, x1) ----------
__global__ void uline_band(float* __restrict__ out, const int* __restrict__ ws) {
    const int b  = blockIdx.x;
    const int y1 = ws[4 * b + 0];
    if (y1 < 0) return;                         // any_black == false
    const int x0 = ws[4 * b + 1];
    const int x1 = ws[4 * b + 2];
    const int ncols = x1 - x0;                  // range(x0, x1) excludes x1
    if (ncols <= 0) return;
    const int r0    = (y1 - 2 < 0) ? 0 : (y1 - 2);
    const int nrows = y1 - r0 + 1;
    const int per_c = nrows * ncols;
    const int total = 3 * per_c;
    const size_t base = (size_t)b * 3 * PLANE_;

    for (int i = threadIdx.x; i < total; i += blockDim.x) {
        const int c   = i / per_c;
        const int rem = i - c * per_c;
        const int ry  = rem / ncols;
        const int rx  = rem - ry * ncols;
        out[base + (size_t)c * PLANE_ + (size_t)(r0 + ry) * W_ + x0 + rx] = 0.0f;
    }
}

extern "C" void kernel_launch(void* const* d_in, const int* in_sizes, int n_in,
                              void* d_out, int out_size, void* d_ws, size_t ws_size,
                              hipStream_t stream) {
    const float* in  = (const float*)d_in[0];
    float*       out = (float*)d_out;
    int*         ws  = (int*)d_ws;
    const int B = in_sizes[0] / (3 * PLANE_);   // 64

    uline_init_ws<<<(B + 255) / 256, 256, 0, stream>>>(ws, B);

    dim3 grid(W_ / TILE_PX, H_, B);             // (2, 128, 64)
    uline_bbox_copy<<<grid, 256, 0, stream>>>(in, out, ws);

    uline_band<<<B, 256, 0, stream>>>(out, ws);
}